// ChebNetII_14164802142902
// MI455X (gfx1250) — compile-verified
//
#include <hip/hip_runtime.h>

#define F_IN 256
#define HID  64
#define KORD 10   // Chebyshev order K

typedef __attribute__((ext_vector_type(2))) float v2f;
typedef __attribute__((ext_vector_type(8))) float v8f;

// ---------------------------------------------------------------------------
// Fused MLP: h = relu(x@W1 + b1) @ W2 + b2, using v_wmma_f32_16x16x4_f32.
// Block = 128 threads (4 wave32), one 16-row tile of x per block.
// Wave w owns output columns [16w, 16w+16). GEMM1 intermediate staged in LDS.
//
// f32 WMMA wave32 layouts (ISA 7.12.2):
//   A 16x4 : lane l<16 holds {A[l][k],A[l][k+1]}, lane l>=16 holds K+2 pair
//   B 4x16 : lanes 0-15 rows K={0,1}, lanes 16-31 rows K={2,3}, N = lane%16
//   C/D    : vgpr i -> M = i + 8*(lane>=16), N = lane%16
// ---------------------------------------------------------------------------
__global__ __launch_bounds__(128)
void mlp_wmma_kernel(const float* __restrict__ x,
                     const float* __restrict__ W1, const float* __restrict__ b1,
                     const float* __restrict__ W2, const float* __restrict__ b2,
                     float* __restrict__ h, int n_rows) {
  __shared__ float lds_h1[16][HID + 4];
  const int wave = threadIdx.x >> 5;
  const int lane = threadIdx.x & 31;
  const int lm   = lane & 15;   // M for A, N for B/C/D
  const int lh   = lane >> 4;   // which K-pair half
  const int row0 = blockIdx.x * 16;
  const int col0 = wave * 16;

  int rr = row0 + lm;
  if (rr >= n_rows) rr = n_rows - 1;          // clamp (keeps EXEC uniform)
  const float* __restrict__ xrow = x + (size_t)rr * F_IN;

  // ---- GEMM1: [16 x 256] @ [256 x 16], K-step 4 ----
  v8f acc;
  {
    float bias = b1[col0 + lm];               // same N for all 8 C elems
    #pragma unroll
    for (int i = 0; i < 8; ++i) acc[i] = bias;
  }
  for (int kb = 0; kb < F_IN; kb += 4) {
    const int k = kb + 2 * lh;
    v2f a; a.x = xrow[k];                    a.y = xrow[k + 1];
    v2f b; b.x = W1[(size_t)k * HID + col0 + lm];
           b.y = W1[(size_t)(k + 1) * HID + col0 + lm];
    acc = __builtin_amdgcn_wmma_f32_16x16x4_f32(false, a, false, b,
                                                (short)0, acc, false, false);
  }
  // relu, stash 16x16 sub-tile into the shared 16x64 h1 tile
  #pragma unroll
  for (int i = 0; i < 8; ++i) {
    const int m = i + 8 * lh;
    const float v = acc[i];
    lds_h1[m][col0 + lm] = v > 0.0f ? v : 0.0f;
  }
  __syncthreads();

  // ---- GEMM2: [16 x 64] @ [64 x 16], K-step 4 ----
  v8f acc2;
  {
    float bias = b2[col0 + lm];
    #pragma unroll
    for (int i = 0; i < 8; ++i) acc2[i] = bias;
  }
  for (int kb = 0; kb < HID; kb += 4) {
    const int k = kb + 2 * lh;
    v2f a; a.x = lds_h1[lm][k];              a.y = lds_h1[lm][k + 1];
    v2f b; b.x = W2[(size_t)k * HID + col0 + lm];
           b.y = W2[(size_t)(k + 1) * HID + col0 + lm];
    acc2 = __builtin_amdgcn_wmma_f32_16x16x4_f32(false, a, false, b,
                                                 (short)0, acc2, false, false);
  }
  #pragma unroll
  for (int i = 0; i < 8; ++i) {
    const int m = i + 8 * lh;
    const int r = row0 + m;
    if (r < n_rows) h[(size_t)r * HID + col0 + lm] = acc2[i];
  }
}

// ---------------------------------------------------------------------------
// Chebyshev coefficients: coe[i] = 2/(K+1) * sum_j temp[j] * T_i(x_j)
// ---------------------------------------------------------------------------
__global__ void coe_kernel(const float* __restrict__ temp, float* __restrict__ coe) {
  const int i = threadIdx.x;
  if (i > KORD) return;
  const double PI = 3.14159265358979323846;
  double s = 0.0;
  for (int j = 0; j <= KORD; ++j) {
    const double xj = cos((double)(KORD - j) + 0.5) == 0.0 ? 0.0 : 0.0; // placeholder never used
    (void)xj;
    const double xx = cos(((double)(KORD - j) + 0.5) * PI / (double)(KORD + 1));
    double t0 = 1.0, t1 = xx, ti = 1.0;
    if (i == 0)      ti = t0;
    else if (i == 1) ti = t1;
    else {
      for (int q = 2; q <= i; ++q) { ti = 2.0 * xx * t1 - t0; t0 = t1; t1 = ti; }
    }
    s += (double)temp[j] * ti;
  }
  coe[i] = (float)((2.0 / (double)(KORD + 1)) * s);
}

__global__ void zero_kernel(float* __restrict__ p, unsigned long long n) {
  unsigned long long i = (unsigned long long)blockIdx.x * blockDim.x + threadIdx.x;
  if (i < n) p[i] = 0.0f;
}

__global__ void deg_kernel(const int* __restrict__ ei, float* __restrict__ deg, int E) {
  int e = blockIdx.x * blockDim.x + threadIdx.x;
  if (e < E) atomicAdd(&deg[ei[e]], 1.0f);          // segment over row = ei[0][e]
}

__global__ void dis_kernel(float* __restrict__ deg, int n) {
  int i = blockIdx.x * blockDim.x + threadIdx.x;
  if (i < n) {
    float d = deg[i];
    deg[i] = d > 0.0f ? rsqrtf(d) : 0.0f;           // in-place deg -> dis
  }
}

__global__ void ew_kernel(const int* __restrict__ ei, const float* __restrict__ dis,
                          float* __restrict__ w, int E) {
  int e = blockIdx.x * blockDim.x + threadIdx.x;
  if (e < E) w[e] = -(dis[ei[e]] * dis[ei[E + e]]);
}

// out[col[e]] += w[e] * v[row[e]] ; 64 lanes per edge (one per feature)
__global__ void prop_kernel(const int* __restrict__ ei, const float* __restrict__ w,
                            const float* __restrict__ v, float* __restrict__ out, int E) {
  unsigned long long t = (unsigned long long)blockIdx.x * blockDim.x + threadIdx.x;
  int e = (int)(t >> 6);
  int f = (int)(t & 63);
  if (e < E) {
    const int src = ei[e];
    const int dst = ei[E + e];
    atomicAdd(&out[(size_t)dst * HID + f], w[e] * v[(size_t)src * HID + f]);
  }
}

__global__ void combine_init_kernel(const float* __restrict__ T0, const float* __restrict__ T1,
                                    const float* __restrict__ coe, float* __restrict__ out,
                                    unsigned long long n) {
  unsigned long long i = (unsigned long long)blockIdx.x * blockDim.x + threadIdx.x;
  if (i < n) out[i] = 0.5f * coe[0] * T0[i] + coe[1] * T1[i];
}

// Tnew = 2*P - Told  (P already in Tnew); out += coe[ci] * Tnew
__global__ void cheb_update_kernel(float* __restrict__ Tnew, const float* __restrict__ Told,
                                   const float* __restrict__ coe, int ci,
                                   float* __restrict__ out, unsigned long long n) {
  unsigned long long i = (unsigned long long)blockIdx.x * blockDim.x + threadIdx.x;
  if (i < n) {
    float t = 2.0f * Tnew[i] - Told[i];
    Tnew[i] = t;
    out[i] += coe[ci] * t;
  }
}

extern "C" void kernel_launch(void* const* d_in, const int* in_sizes, int n_in,
                              void* d_out, int out_size, void* d_ws, size_t ws_size,
                              hipStream_t stream) {
  const float* x    = (const float*)d_in[0];
  const int*   ei   = (const int*)d_in[1];   // [2, E] row-major: rows then cols
  const float* W1   = (const float*)d_in[2];
  const float* b1   = (const float*)d_in[3];
  const float* W2   = (const float*)d_in[4];
  const float* b2   = (const float*)d_in[5];
  const float* temp = (const float*)d_in[6];
  float* out = (float*)d_out;

  const int N = in_sizes[0] / F_IN;
  const int E = in_sizes[1] / 2;
  const unsigned long long nh = (unsigned long long)N * HID;

  // workspace layout: T0 | T1 | T2 | w[E] | dis[N] | coe[16]
  float* ws   = (float*)d_ws;
  float* T[3] = { ws, ws + nh, ws + 2 * nh };
  float* wbuf = ws + 3 * nh;
  float* dis  = wbuf + E;
  float* coe  = dis + N;

  const int TB = 256;
  const unsigned gN   = (unsigned)((N + TB - 1) / TB);
  const unsigned gE   = (unsigned)((E + TB - 1) / TB);
  const unsigned gNH  = (unsigned)((nh + TB - 1) / TB);
  const unsigned gPr  = (unsigned)(((unsigned long long)E * 64ull + TB - 1) / TB);
  const unsigned gMlp = (unsigned)((N + 15) / 16);

  // coefficients + graph prep
  coe_kernel<<<1, 32, 0, stream>>>(temp, coe);
  zero_kernel<<<gN, TB, 0, stream>>>(dis, (unsigned long long)N);
  deg_kernel<<<gE, TB, 0, stream>>>(ei, dis, E);
  dis_kernel<<<gN, TB, 0, stream>>>(dis, N);
  ew_kernel<<<gE, TB, 0, stream>>>(ei, dis, wbuf, E);

  // MLP -> T0  (Tx_0 = h)
  mlp_wmma_kernel<<<gMlp, 128, 0, stream>>>(x, W1, b1, W2, b2, T[0], N);

  // Tx_1 = prop(h)
  zero_kernel<<<gNH, TB, 0, stream>>>(T[1], nh);
  prop_kernel<<<gPr, TB, 0, stream>>>(ei, wbuf, T[0], T[1], E);

  // out = coe0/2 * Tx0 + coe1 * Tx1
  combine_init_kernel<<<gNH, TB, 0, stream>>>(T[0], T[1], coe, out, nh);

  // Chebyshev recurrence, fixed deterministic buffer rotation
  int a = 0, b = 1, c = 2;
  for (int i = 2; i <= KORD; ++i) {
    zero_kernel<<<gNH, TB, 0, stream>>>(T[c], nh);
    prop_kernel<<<gPr, TB, 0, stream>>>(ei, wbuf, T[b], T[c], E);
    cheb_update_kernel<<<gNH, TB, 0, stream>>>(T[c], T[a], coe, i, out, nh);
    const int t = a; a = b; b = c; c = t;
  }
}